// Dynamics_72877005079077
// MI455X (gfx1250) — compile-verified
//
#include <hip/hip_runtime.h>
#include <hip/hip_bf16.h>

typedef float v2f __attribute__((ext_vector_type(2)));
typedef float v8f __attribute__((ext_vector_type(8)));

static __device__ __forceinline__ v8f wmma4(v2f a, v2f b, v8f c) {
    // D = A(16x4,f32) * B(4x16,f32) + C(16x16,f32)
    return __builtin_amdgcn_wmma_f32_16x16x4_f32(false, a, false, b, (short)0, c, false, false);
}

// Branch-free tanh: tanh(z) = sign(z) * (1 - 2/(1 + exp2(2*log2(e)*|z|)))
// v_exp_f32 + v_rcp_f32 are ~1 ulp; |z|->inf gives rcp(inf)=0 -> exactly +-1.
static __device__ __forceinline__ float fast_tanh(float z) {
    float az = __builtin_fabsf(z);
    float e  = __builtin_amdgcn_exp2f(az * 2.8853900817779268f); // 2/ln(2)
    float r  = __builtin_amdgcn_rcpf(e + 1.0f);
    float t  = __builtin_fmaf(-2.0f, r, 1.0f);
    return __builtin_copysignf(t, z);
}

#define WAVES 8
#define TILE  16   // samples per wave

__global__ __launch_bounds__(256) void Dynamics_72877005079077_kernel(
    const float* __restrict__ X,      // (B,16): [x | v]
    const float* __restrict__ W1,     // (8,64)
    const float* __restrict__ B1,     // (64)
    const float* __restrict__ W2,     // (64,1)
    const float* __restrict__ Kmat,   // (8,8)
    const float* __restrict__ Dmat,   // (8,8)
    const float* __restrict__ Attr,   // (8)
    float* __restrict__ out)          // (B,8)
{
    __shared__ float sW1[16 * 66];          // padded W1, rows 8..15 zero
    __shared__ float sB1[64];
    __shared__ float sW2[64];
    __shared__ float sU[WAVES][16 * 66];    // u = w2*s, per-wave [sample][hidden]
    __shared__ float sG[WAVES][16 * 9];     // g vectors, padded rows
    __shared__ float sVHV[WAVES][16];       // v^T H v per sample

    const int tid  = threadIdx.x;
    const int wave = tid >> 5;
    const int lane = tid & 31;
    const int lh   = lane & 15;
    const int hi   = lane >> 4;
    const int sbase = blockIdx.x * (WAVES * TILE) + wave * TILE;

    // ---- preload W1 (padded, zero-filled), b1, w2 ----
    for (int idx = tid; idx < 16 * 66; idx += 256) {
        int row = idx / 66, col = idx - row * 66;
        sW1[idx] = (row < 8 && col < 64) ? W1[row * 64 + col] : 0.0f;
    }
    if (tid < 64) { sB1[tid] = B1[tid]; sW2[tid] = W2[tid]; }
    __syncthreads();

    // ---- A fragments for Z = x@W1 and P = v@W1 (A: 16x4 f32 layout) ----
    const float* xrow = X + (size_t)(sbase + lh) * 16;
    v2f ax0 = *(const v2f*)(xrow + 0 + hi * 2);       // K 0..3 half
    v2f ax1 = *(const v2f*)(xrow + 4 + hi * 2);       // K 4..7 half
    v2f av0 = *(const v2f*)(xrow + 8 + hi * 2);
    v2f av1 = *(const v2f*)(xrow + 12 + hi * 2);

    const v8f zero8 = {0.f,0.f,0.f,0.f,0.f,0.f,0.f,0.f};
    float vhv[8];
#pragma unroll
    for (int r = 0; r < 8; ++r) vhv[r] = 0.0f;

    // ---- Z/P tiles over 4 N-tiles of the 64 hidden units ----
#pragma unroll
    for (int n = 0; n < 4; ++n) {
        const int col = n * 16 + lh;
        v2f b0, b1f;   // B: 4x16 f32 layout (K = hi*2 + vgpr)
        b0.x  = sW1[(0 + hi * 2) * 66 + col];
        b0.y  = sW1[(1 + hi * 2) * 66 + col];
        b1f.x = sW1[(4 + hi * 2) * 66 + col];
        b1f.y = sW1[(5 + hi * 2) * 66 + col];

        v8f z = wmma4(ax1, b1f, wmma4(ax0, b0, zero8));
        v8f p = wmma4(av1, b1f, wmma4(av0, b0, zero8));

        const float bb = sB1[col];
        const float ww = sW2[col];
#pragma unroll
        for (int r = 0; r < 8; ++r) {
            float zz = z[r] + bb;
            float t  = fast_tanh(zz);
            float s  = 1.0f - t * t;
            float u  = ww * s;             // u_h = w2_h * s_h
            float c  = -2.0f * t * u;      // c_h = -2 t s w2
            float pv = p[r];
            vhv[r] += c * pv * pv;
            sU[wave][(r + hi * 8) * 66 + col] = u;
        }
    }

    // ---- reduce v^T H v across the 16 lanes of each half-wave ----
#pragma unroll
    for (int r = 0; r < 8; ++r) {
        float s = vhv[r];
        s += __shfl_xor(s, 1, 32);
        s += __shfl_xor(s, 2, 32);
        s += __shfl_xor(s, 4, 32);
        s += __shfl_xor(s, 8, 32);
        if (lh == 0) sVHV[wave][hi * 8 + r] = s;
    }
    __syncthreads();

    // ---- G = U @ W1^T  (K = 64, 16 chained WMMAs; B cols 8..15 are zero) ----
    v8f g = zero8;
    const float* uRow = &sU[wave][lh * 66];
    const float* wRow = &sW1[lh * 66];
#pragma unroll
    for (int kk = 0; kk < 16; ++kk) {
        const int k0 = kk * 4 + hi * 2;
        v2f au = *(const v2f*)(uRow + k0);
        v2f bu = *(const v2f*)(wRow + k0);
        g = wmma4(au, bu, g);
    }
    if (lh < 8) {
#pragma unroll
        for (int r = 0; r < 8; ++r) sG[wave][(r + hi * 8) * 9 + lh] = g[r];
    }
    __syncthreads();

    // ---- final per-sample combine: lanes 0..15 own one sample each ----
    if (lane < 16) {
        const int m = sbase + lane;
        const float* xr = X + (size_t)m * 16;
        float x[8], vv[8], gvec[8], xa[8], force[8];
#pragma unroll
        for (int j = 0; j < 8; ++j) { x[j] = xr[j]; vv[j] = xr[8 + j]; }
#pragma unroll
        for (int i = 0; i < 8; ++i) gvec[i] = sG[wave][lane * 9 + i];

        float gg = 0.0f;
#pragma unroll
        for (int i = 0; i < 8; ++i) gg += gvec[i] * gvec[i];
#pragma unroll
        for (int j = 0; j < 8; ++j) xa[j] = x[j] - Attr[j];
#pragma unroll
        for (int q = 0; q < 8; ++q) {
            float f = 0.0f;
#pragma unroll
            for (int j = 0; j < 8; ++j)
                f += vv[j] * Dmat[j * 8 + q] + xa[j] * Kmat[j * 8 + q];
            force[q] = -f;
        }
        float gf = 0.0f;
#pragma unroll
        for (int q = 0; q < 8; ++q) gf += gvec[q] * force[q];

        const float scale = (gf + sVHV[wave][lane]) / (1.0f + gg);
        float* orow = out + (size_t)m * 8;
#pragma unroll
        for (int q = 0; q < 8; ++q) orow[q] = force[q] - gvec[q] * scale;
    }
}

extern "C" void kernel_launch(void* const* d_in, const int* in_sizes, int n_in,
                              void* d_out, int out_size, void* d_ws, size_t ws_size,
                              hipStream_t stream) {
    const float* X    = (const float*)d_in[0];
    const float* W1   = (const float*)d_in[1];
    const float* B1   = (const float*)d_in[2];
    const float* W2   = (const float*)d_in[3];
    // d_in[4] = b2: does not affect the output (only shifts the embedding value)
    const float* Kmat = (const float*)d_in[5];
    const float* Dmat = (const float*)d_in[6];
    const float* Attr = (const float*)d_in[7];
    float* out = (float*)d_out;

    const int Btot   = in_sizes[0] / 16;          // 131072
    const int blocks = Btot / (WAVES * TILE);     // 1024 blocks of 256 threads
    Dynamics_72877005079077_kernel<<<blocks, 256, 0, stream>>>(
        X, W1, B1, W2, Kmat, Dmat, Attr, out);
}